// Cnn_5626407158447
// MI455X (gfx1250) — compile-verified
//
#include <hip/hip_runtime.h>
#include <hip/hip_bf16.h>

typedef __attribute__((ext_vector_type(2))) float v2f;
typedef __attribute__((ext_vector_type(8))) float v8f;

#define RSTRIDE 68      // padded LDS row stride (floats): 4-bank shift/row kills conflicts
#define NPTS    8192
#define FRAGL   3328    // per-layer fragment floats: 13 ksteps * 4 ntiles * 32 lanes * 2

struct WB { const float* W[6]; const float* b[6]; };

// ---------------------------------------------------------------------------
// Pre-swizzle weights into WMMA B-fragment order so a lane's fragment is
// 8 contiguous bytes:  frag[l][kk][nt][lane][2]  (zero-padded to 52x64).
// ws layout (floats): frag[6][FRAGL] | bpad[6][64]
// ---------------------------------------------------------------------------
__global__ void pack_weights(WB wb, float* ws) {
    const int in_d[6]  = {2, 50, 50, 50, 50, 50};
    const int out_d[6] = {50, 50, 50, 50, 50, 2};
    float* frag = ws;
    float* bp   = ws + 6 * FRAGL;
    for (int i = threadIdx.x; i < 6 * 64; i += blockDim.x) {
        const int l = i >> 6, n = i & 63;
        bp[i] = (n < out_d[l]) ? wb.b[l][n] : 0.f;
    }
    for (int i = threadIdx.x; i < 6 * FRAGL; i += blockDim.x) {
        const int l    = i / FRAGL;
        const int r    = i % FRAGL;
        const int kk   = r / 256;        // k-step 0..12
        const int nt   = (r % 256) / 64; // n-tile 0..3
        const int r3   = r % 64;
        const int lane = r3 >> 1;
        const int half = r3 & 1;
        const int n = nt * 16 + (lane & 15);
        const int k = kk * 4 + ((lane >> 4) << 1) + half;
        float v = 0.f;
        if (k < in_d[l] && n < out_d[l]) v = wb.W[l][k * out_d[l] + n];
        frag[i] = v;
    }
}

// ---------------------------------------------------------------------------
// One wave per 16-point tile. Propagates 4 channels (value, d/dx, d/dt,
// d2/dt2) through the MLP with V_WMMA_F32_16X16X4_F32. Per layer: stage the
// 13.3KB swizzled weight block into LDS once (b128 coalesced), reuse it for
// all 4 channels via conflict-free ds_load_b64; prefetch next layer's block.
// ---------------------------------------------------------------------------
__global__ void __launch_bounds__(32) mlp_wmma(const float* __restrict__ x,
                                               const float* __restrict__ wsw,
                                               float* __restrict__ outbuf) {
    __shared__ float act[4][16 * RSTRIDE];
    __shared__ float zbuf[4][16 * RSTRIDE];
    __shared__ float wfrag[FRAGL];

    const int lane = threadIdx.x;
    const int m0   = blockIdx.x * 16;
    const float* frag = wsw;
    const float* bp   = wsw + 6 * FRAGL;

    // zero activations (padded columns stay zero through all layers)
    float* ap = &act[0][0];
    for (int i = lane; i < 4 * 16 * RSTRIDE; i += 32) ap[i] = 0.f;
    __syncthreads();
    if (lane < 16) {
        const int r = lane;
        act[0][r * RSTRIDE + 0] = x[(m0 + r) * 2 + 0];  // x-coord
        act[0][r * RSTRIDE + 1] = x[(m0 + r) * 2 + 1];  // t-coord
        act[1][r * RSTRIDE + 0] = 1.f;                  // d/dx seed
        act[2][r * RSTRIDE + 1] = 1.f;                  // d/dt seed
    }
    __syncthreads();

    const int row   = lane & 15;            // A-fragment row
    const int koff  = (lane >> 4) << 1;     // A fragment K sub-offset: 0 or 2
    const int ncol  = lane & 15;            // C/D column within a 16-wide tile
    const int mbase = (lane < 16) ? 0 : 8;  // C/D row base

    for (int l = 0; l < 6; ++l) {
        // ---- stage this layer's swizzled weight fragments into LDS (b128) ----
        const float* gf = frag + l * FRAGL;
        const int nload = (l == 0) ? 64 : (FRAGL / 4);  // layer 0 uses only kk=0
        for (int i = lane; i < nload; i += 32)
            ((float4*)wfrag)[i] = ((const float4*)gf)[i];
        if (l + 1 < 6)  // overlap next layer's L2 fetch with this layer's WMMAs
            __builtin_prefetch(frag + (l + 1) * FRAGL + lane * 104, 0, 0);
        __syncthreads();

        const int kmax = (l == 0) ? 1 : 13;

        for (int c = 0; c < 4; ++c) {
            v8f acc[4];
            for (int nt = 0; nt < 4; ++nt) {
                const float binit = (c == 0) ? bp[l * 64 + nt * 16 + ncol] : 0.f;
                for (int j = 0; j < 8; ++j) acc[nt][j] = binit;
            }
            for (int kk = 0; kk < kmax; ++kk) {
                const int k0 = kk * 4;
                v2f a;
                a.x = act[c][row * RSTRIDE + k0 + koff];
                a.y = act[c][row * RSTRIDE + k0 + koff + 1];
                for (int nt = 0; nt < 4; ++nt) {
                    const v2f b = *(const v2f*)&wfrag[kk * 256 + nt * 64 + (lane << 1)];
                    acc[nt] = __builtin_amdgcn_wmma_f32_16x16x4_f32(
                        false, a, false, b, (short)0, acc[nt], false, false);
                }
            }
            for (int nt = 0; nt < 4; ++nt)
                for (int j = 0; j < 8; ++j)
                    zbuf[c][(mbase + j) * RSTRIDE + nt * 16 + ncol] = acc[nt][j];
        }
        __syncthreads();

        if (l < 5) {
            // tanh + exact forward-mode derivative propagation
            for (int t = 0; t < 32; ++t) {
                const int idx = lane + t * 32;
                const int m = idx >> 6, n = idx & 63;
                const float z   = zbuf[0][m * RSTRIDE + n];
                const float zx  = zbuf[1][m * RSTRIDE + n];
                const float zt  = zbuf[2][m * RSTRIDE + n];
                const float ztt = zbuf[3][m * RSTRIDE + n];
                const float h = tanhf(z);
                const float s = 1.f - h * h;
                act[0][m * RSTRIDE + n] = h;
                act[1][m * RSTRIDE + n] = s * zx;
                act[2][m * RSTRIDE + n] = s * zt;
                act[3][m * RSTRIDE + n] = s * ztt - 2.f * h * s * zt * zt;
            }
            __syncthreads();
        }
    }

    if (lane < 16) {
        const int r = lane, i = m0 + r;
        const float u   = zbuf[0][r * RSTRIDE + 0];
        const float v   = zbuf[0][r * RSTRIDE + 1];
        const float ux  = zbuf[1][r * RSTRIDE + 0];
        const float vx  = zbuf[1][r * RSTRIDE + 1];
        const float utt = zbuf[3][r * RSTRIDE + 0];
        const float vtt = zbuf[3][r * RSTRIDE + 1];
        const float Q = u * u + v * v;
        outbuf[0 * NPTS + i] = -vx;     // c1
        outbuf[1 * NPTS + i] =  ux;     // c2
        outbuf[2 * NPTS + i] =  utt;    // d1
        outbuf[3 * NPTS + i] =  vtt;    // d2
        outbuf[4 * NPTS + i] =  Q * u;  // e1
        outbuf[5 * NPTS + i] =  Q * v;  // e2
    }
}

// ---------------------------------------------------------------------------
// Six deterministic tree reductions over N (no float atomics -> bitwise
// reproducible across graph replays).  sums = {Scc,Sdd,See,Scd,Sce,Sde}
// ---------------------------------------------------------------------------
__global__ void __launch_bounds__(256) reduce6(const float* __restrict__ buf,
                                               float* __restrict__ sums) {
    __shared__ float red[256][6];
    const int tid = threadIdx.x;
    float s0 = 0.f, s1 = 0.f, s2 = 0.f, s3 = 0.f, s4 = 0.f, s5 = 0.f;
    for (int i = tid; i < NPTS; i += 256) {
        const float c1 = buf[0 * NPTS + i], c2 = buf[1 * NPTS + i];
        const float d1 = buf[2 * NPTS + i], d2 = buf[3 * NPTS + i];
        const float e1 = buf[4 * NPTS + i], e2 = buf[5 * NPTS + i];
        s0 += c1 * c1 + c2 * c2;
        s1 += d1 * d1 + d2 * d2;
        s2 += e1 * e1 + e2 * e2;
        s3 += c1 * d1 + c2 * d2;
        s4 += c1 * e1 + c2 * e2;
        s5 += d1 * e1 + d2 * e2;
    }
    red[tid][0] = s0; red[tid][1] = s1; red[tid][2] = s2;
    red[tid][3] = s3; red[tid][4] = s4; red[tid][5] = s5;
    __syncthreads();
    for (int off = 128; off > 0; off >>= 1) {
        if (tid < off)
            for (int j = 0; j < 6; ++j) red[tid][j] += red[tid + off][j];
        __syncthreads();
    }
    if (tid == 0)
        for (int j = 0; j < 6; ++j) sums[j] = red[0][j];
}

// ---------------------------------------------------------------------------
// out[p] = (Scc + b^2 Sdd + a^2 See + 2b Scd + 2a Sce + 2ab Sde) / N
// ---------------------------------------------------------------------------
__global__ void finalize(const float* __restrict__ para,
                         const float* __restrict__ sums,
                         float* __restrict__ out, int P) {
    const int p = blockIdx.x * blockDim.x + threadIdx.x;
    if (p >= P) return;
    const float a = para[p * 3 + 0];
    const float b = para[p * 3 + 1];
    const float Scc = sums[0], Sdd = sums[1], See = sums[2];
    const float Scd = sums[3], Sce = sums[4], Sde = sums[5];
    out[p] = (Scc + b * b * Sdd + a * a * See +
              2.f * b * Scd + 2.f * a * Sce + 2.f * a * b * Sde) * (1.0f / (float)NPTS);
}

extern "C" void kernel_launch(void* const* d_in, const int* in_sizes, int n_in,
                              void* d_out, int out_size, void* d_ws, size_t ws_size,
                              hipStream_t stream) {
    const float* x    = (const float*)d_in[0];
    const float* para = (const float*)d_in[1];
    WB wb;
    for (int l = 0; l < 6; ++l) {
        wb.W[l] = (const float*)d_in[2 + l];
        wb.b[l] = (const float*)d_in[8 + l];
    }
    float* ws     = (float*)d_ws;
    float* outbuf = ws + 6 * FRAGL + 6 * 64;  // 6 x 8192 floats
    float* sums   = outbuf + 6 * NPTS;        // 6 floats

    pack_weights<<<1, 256, 0, stream>>>(wb, ws);
    mlp_wmma<<<NPTS / 16, 32, 0, stream>>>(x, ws, outbuf);
    reduce6<<<1, 256, 0, stream>>>(outbuf, sums);
    const int P = in_sizes[1] / 3;
    finalize<<<(P + 255) / 256, 256, 0, stream>>>(para, sums, (float*)d_out, P);
}